// NodeEmbedding_90606630076672
// MI455X (gfx1250) — compile-verified
//
#include <hip/hip_runtime.h>
#include <hip/hip_bf16.h>

#define N_NODES 100000
#define N_EDGES 1600000
#define D 128
#define BN_EPS 1e-5f

typedef __attribute__((ext_vector_type(2))) float v2f;
typedef __attribute__((ext_vector_type(8))) float v8f;

__device__ __forceinline__ void atomic_add_f32(float* p, float v) {
    __hip_atomic_fetch_add(p, v, __ATOMIC_RELAXED, __HIP_MEMORY_SCOPE_AGENT);
}

// ---------------------------------------------------------------------------
// K0: init deg=1.0 (self loop), stats=0
// ---------------------------------------------------------------------------
__global__ void gcn_init(float* __restrict__ deg, float* __restrict__ stats) {
    int i = blockIdx.x * blockDim.x + threadIdx.x;
    if (i < N_NODES) deg[i] = 1.0f;
    if (i < 256)     stats[i] = 0.0f;
}

// ---------------------------------------------------------------------------
// K1: degree accumulation (non-self edges), deg over source (row) index
// ---------------------------------------------------------------------------
__global__ void gcn_degree(const long long* __restrict__ ei, float* __restrict__ deg) {
    int e = blockIdx.x * blockDim.x + threadIdx.x;
    if (e >= N_EDGES) return;
    long long row = ei[e];
    long long col = ei[N_EDGES + e];
    if (row != col) atomic_add_f32(&deg[row], 1.0f);
}

// ---------------------------------------------------------------------------
// K2: h = x @ W using fp32 WMMA (V_WMMA_F32_16X16X4_F32).
// Block = 256 threads = 8 waves. Block covers 64 rows (4 m-tiles of 16),
// wave w covers columns [16w, 16w+16). x-tile staged in LDS (32 KB) and
// shared across all 8 waves; each B fragment is reused by 4 accumulators.
// 100000 % 64 == 32, so only the last block takes the guarded tail path.
// ---------------------------------------------------------------------------
__global__ void __launch_bounds__(256) gcn_gemm(const float* __restrict__ X,
                                                const float* __restrict__ W,
                                                float* __restrict__ H) {
    __shared__ __align__(16) float sA[64 * D];  // 32 KB

    const int tid = threadIdx.x;
    const int m0  = blockIdx.x * 64;

    // Cooperative load of 64 rows into LDS as float4 (row-clamped for the tail
    // block so every lane stays active — WMMA requires EXEC == all-1s).
    for (int i = tid; i < 64 * (D / 4); i += 256) {
        int row  = m0 + (i >> 5);
        int srow = row < N_NODES ? row : (N_NODES - 1);
        ((float4*)sA)[i] = ((const float4*)X)[(size_t)srow * (D / 4) + (i & 31)];
    }
    __syncthreads();

    const int wave = tid >> 5;
    const int lane = tid & 31;
    const int n0   = wave * 16;
    const int half = lane >> 4;   // 0: K pair {0,1} / rows 0..7 ; 1: K pair {2,3} / rows 8..15
    const int r    = lane & 15;

    v8f acc0 = {}, acc1 = {}, acc2 = {}, acc3 = {};

    for (int k0 = 0; k0 < D; k0 += 4) {
        const int kA = k0 + 2 * half;
        // B fragment (4x16): lanes 0-15 hold rows K=kA,kA+1; lanes 16-31 K=kA+2,kA+3
        // (the +2 is folded into kA via `half`).
        v2f b;
        b.x = W[(size_t)kA * D + n0 + r];
        b.y = W[(size_t)(kA + 1) * D + n0 + r];
        // A fragments: contiguous float2 per lane from LDS (kA even -> 8B aligned)
        v2f a0 = *(const v2f*)&sA[(r) * D + kA];
        v2f a1 = *(const v2f*)&sA[(16 + r) * D + kA];
        v2f a2 = *(const v2f*)&sA[(32 + r) * D + kA];
        v2f a3 = *(const v2f*)&sA[(48 + r) * D + kA];
        acc0 = __builtin_amdgcn_wmma_f32_16x16x4_f32(false, a0, false, b, (short)0, acc0, false, false);
        acc1 = __builtin_amdgcn_wmma_f32_16x16x4_f32(false, a1, false, b, (short)0, acc1, false, false);
        acc2 = __builtin_amdgcn_wmma_f32_16x16x4_f32(false, a2, false, b, (short)0, acc2, false, false);
        acc3 = __builtin_amdgcn_wmma_f32_16x16x4_f32(false, a3, false, b, (short)0, acc3, false, false);
    }

    // C/D layout: VGPR v holds M = v (lanes 0-15) or M = v+8 (lanes 16-31), N = r.
    const int cbase = 8 * half;
    if (m0 + 64 <= N_NODES) {
        // Fast path: full tile, no guards (1562 of 1563 blocks).
        float* p0 = H + (size_t)(m0 + cbase) * D + n0 + r;
#pragma unroll
        for (int v = 0; v < 8; ++v) {
            p0[(size_t)(v) * D]      = acc0[v];
            p0[(size_t)(v + 16) * D] = acc1[v];
            p0[(size_t)(v + 32) * D] = acc2[v];
            p0[(size_t)(v + 48) * D] = acc3[v];
        }
    } else {
#pragma unroll
        for (int v = 0; v < 8; ++v) {
            int rr = v + cbase;
            int row0 = m0 + rr;
            int row1 = m0 + 16 + rr;
            int row2 = m0 + 32 + rr;
            int row3 = m0 + 48 + rr;
            if (row0 < N_NODES) H[(size_t)row0 * D + n0 + r] = acc0[v];
            if (row1 < N_NODES) H[(size_t)row1 * D + n0 + r] = acc1[v];
            if (row2 < N_NODES) H[(size_t)row2 * D + n0 + r] = acc2[v];
            if (row3 < N_NODES) H[(size_t)row3 * D + n0 + r] = acc3[v];
        }
    }
}

// ---------------------------------------------------------------------------
// K3: dis = rsqrt(deg); out = dis^2 * h  (self-loop contribution, also
// initializes the whole aggregation buffer so no separate zeroing pass)
// ---------------------------------------------------------------------------
__global__ void gcn_self(const float* __restrict__ deg, float* __restrict__ dis,
                         const float* __restrict__ H, float* __restrict__ out) {
    int idx = blockIdx.x * blockDim.x + threadIdx.x;   // over N_NODES * 32 float4s
    if (idx >= N_NODES * (D / 4)) return;
    int node = idx >> 5;
    float d = rsqrtf(deg[node]);       // deg >= 1 always (self loop)
    if ((idx & 31) == 0) dis[node] = d;
    float s = d * d;                   // = 1/deg
    float4 v = ((const float4*)H)[idx];
    float4 o = make_float4(s * v.x, s * v.y, s * v.z, s * v.w);
    ((float4*)out)[idx] = o;
}

// ---------------------------------------------------------------------------
// K4: edge scatter. One wave32 per edge: 32 lanes x float4 = one 128-wide row.
// out[col] += dis[row]*dis[col] * h[row]  via hardware f32 atomics (L2).
// ---------------------------------------------------------------------------
__global__ void __launch_bounds__(256) gcn_scatter(const long long* __restrict__ ei,
                                                   const float* __restrict__ H,
                                                   const float* __restrict__ dis,
                                                   float* __restrict__ out) {
    int warp = (blockIdx.x * blockDim.x + threadIdx.x) >> 5;
    int lane = threadIdx.x & 31;
    if (warp >= N_EDGES) return;
    long long row = ei[warp];
    long long col = ei[N_EDGES + warp];
    if (row == col) return;                       // remove_self_loops (ew = 0)
    float norm = dis[row] * dis[col];
    float4 v = ((const float4*)(H + (size_t)row * D))[lane];
    float* op = out + (size_t)col * D + lane * 4;
    atomic_add_f32(op + 0, norm * v.x);
    atomic_add_f32(op + 1, norm * v.y);
    atomic_add_f32(op + 2, norm * v.z);
    atomic_add_f32(op + 3, norm * v.w);
}

// ---------------------------------------------------------------------------
// K5: out = relu(out + b); accumulate per-channel sum / sumsq.
// blockDim = 128: thread == channel -> fully coalesced column walk,
// one atomic per channel per block. Full-block fast path (no per-row branch).
// ---------------------------------------------------------------------------
__global__ void __launch_bounds__(128) gcn_relu_stats(float* __restrict__ out,
                                                      const float* __restrict__ b,
                                                      float* __restrict__ stats) {
    int c  = threadIdx.x;            // 0..127
    int r0 = blockIdx.x * 128;
    float bias = b[c];
    float s = 0.0f, ss = 0.0f;
    if (r0 + 128 <= N_NODES) {
#pragma unroll 4
        for (int i = 0; i < 128; ++i) {
            size_t off = (size_t)(r0 + i) * D + c;
            float v = out[off] + bias;
            v = v > 0.0f ? v : 0.0f;
            out[off] = v;
            s  += v;
            ss += v * v;
        }
    } else {
        for (int i = 0; i < 128; ++i) {
            int row = r0 + i;
            if (row >= N_NODES) break;
            size_t off = (size_t)row * D + c;
            float v = out[off] + bias;
            v = v > 0.0f ? v : 0.0f;
            out[off] = v;
            s  += v;
            ss += v * v;
        }
    }
    atomic_add_f32(&stats[c], s);
    atomic_add_f32(&stats[D + c], ss);
}

// ---------------------------------------------------------------------------
// K6: BatchNorm (training: batch stats, biased variance)
// ---------------------------------------------------------------------------
__global__ void gcn_bn(float* __restrict__ out, const float* __restrict__ stats,
                       const float* __restrict__ gamma, const float* __restrict__ beta) {
    size_t idx = (size_t)blockIdx.x * blockDim.x + threadIdx.x;
    if (idx >= (size_t)N_NODES * D) return;
    int c = (int)(idx & (D - 1));
    const float invN = 1.0f / (float)N_NODES;
    float mean = stats[c] * invN;
    float var  = stats[D + c] * invN - mean * mean;
    float inv  = rsqrtf(var + BN_EPS);
    out[idx] = (out[idx] - mean) * inv * gamma[c] + beta[c];
}

// ---------------------------------------------------------------------------
extern "C" void kernel_launch(void* const* d_in, const int* in_sizes, int n_in,
                              void* d_out, int out_size, void* d_ws, size_t ws_size,
                              hipStream_t stream) {
    const float*     x     = (const float*)d_in[0];
    const long long* ei    = (const long long*)d_in[1];   // int64 [2, N_EDGES]
    const float*     W     = (const float*)d_in[2];
    const float*     b     = (const float*)d_in[3];
    const float*     gamma = (const float*)d_in[4];
    const float*     beta  = (const float*)d_in[5];
    float*           out   = (float*)d_out;

    // workspace layout
    float* ws    = (float*)d_ws;
    float* h     = ws;                                 // N_NODES * D
    float* deg   = h + (size_t)N_NODES * D;            // N_NODES
    float* dis   = deg + N_NODES;                      // N_NODES
    float* stats = dis + N_NODES;                      // 256 (sum | sumsq)

    // K0: init
    gcn_init<<<(N_NODES + 255) / 256, 256, 0, stream>>>(deg, stats);
    // K1: degrees
    gcn_degree<<<(N_EDGES + 255) / 256, 256, 0, stream>>>(ei, deg);
    // K2: GEMM h = x @ W  (fp32 WMMA)
    gcn_gemm<<<(N_NODES + 63) / 64, 256, 0, stream>>>(x, W, h);
    // K3: dis + self-loop init of aggregation buffer (d_out)
    gcn_self<<<(N_NODES * (D / 4) + 255) / 256, 256, 0, stream>>>(deg, dis, h, out);
    // K4: edge scatter-add (one wave per edge)
    gcn_scatter<<<(N_EDGES * 32 + 255) / 256, 256, 0, stream>>>(ei, h, dis, out);
    // K5: bias + relu + channel stats
    gcn_relu_stats<<<(N_NODES + 127) / 128, 128, 0, stream>>>(out, b, stats);
    // K6: batchnorm
    gcn_bn<<<(int)(((size_t)N_NODES * D + 255) / 256), 256, 0, stream>>>(out, stats, gamma, beta);
}